// graphsage_max_14250701488886
// MI455X (gfx1250) — compile-verified
//
#include <hip/hip_runtime.h>
#include <hip/hip_bf16.h>

// ---------------------------------------------------------------------------
// GraphSAGE (max-pool aggregation), 5 layers, fp32 end-to-end.
// GEMMs run on the CDNA5 matrix pipe via V_WMMA_F32_16X16X4_F32 (full fp32
// precision -- the workload is scatter/gather bound, not GEMM bound, so there
// is no payoff in dropping to bf16/fp8).
// ---------------------------------------------------------------------------

typedef float v2f __attribute__((ext_vector_type(2)));
typedef float v8f __attribute__((ext_vector_type(8)));

#define NODES   65536
#define LDH     128       // uniform feature-buffer stride (pad cols are zero)

// --- zero fill ---------------------------------------------------------------
__global__ void k_zero(float* __restrict__ p, int n) {
    int i = blockIdx.x * blockDim.x + threadIdx.x;
    if (i < n) p[i] = 0.0f;
}

// --- pad a [rows x cols] row-major matrix into [RP x CP] with zero fill ------
__global__ void k_pad(const float* __restrict__ src, float* __restrict__ dst,
                      int rows, int cols, int RP, int CP) {
    int t = blockIdx.x * blockDim.x + threadIdx.x;
    if (t >= RP * CP) return;
    int r = t / CP, c = t - r * CP;
    dst[t] = (r < rows && c < cols) ? src[r * cols + c] : 0.0f;
}

// --- row L2-normalize x[N,128] -> h[N,128]; one wave32 per node --------------
__global__ void k_normalize(const float* __restrict__ x, float* __restrict__ h,
                            int nNodes) {
    int node = blockIdx.x * (blockDim.x >> 5) + (threadIdx.x >> 5);
    int lane = threadIdx.x & 31;
    if (node >= nNodes) return;
    const float4* xv = (const float4*)(x + (size_t)node * LDH);
    float4 v = xv[lane];                       // lane covers 4 consecutive cols
    float s = v.x * v.x + v.y * v.y + v.z * v.z + v.w * v.w;
    #pragma unroll
    for (int off = 16; off > 0; off >>= 1) s += __shfl_xor(s, off, 32);
    float scale = 1.0f / fmaxf(sqrtf(s), 1e-12f);
    float4 o; o.x = v.x * scale; o.y = v.y * scale;
    o.z = v.z * scale; o.w = v.w * scale;
    ((float4*)(h + (size_t)node * LDH))[lane] = o;
}

// --- edge scatter-max: agg[dst] = max(agg[dst], hp[src]); one wave per edge --
// hp >= 0 (ReLU output) and agg is zero-initialized, so unsigned-int atomicMax
// on the float bit pattern is exact (IEEE order == uint order for x >= 0).
__global__ void k_scatter_max(const float* __restrict__ hp,
                              const int* __restrict__ src,
                              const int* __restrict__ dst,
                              float* __restrict__ agg,
                              int E, int ncols) {
    int gid  = blockIdx.x * blockDim.x + threadIdx.x;
    int e    = gid >> 5;
    int lane = gid & 31;
    if (e >= E) return;
    int s = src[e], d = dst[e];
    const float* hs = hp + (size_t)s * LDH;
    unsigned int* ad = (unsigned int*)(agg + (size_t)d * LDH);
    for (int c = lane; c < ncols; c += 32) {
        float v = hs[c];
        if (v > 0.0f) atomicMax(ad + c, __float_as_uint(v));
    }
}

// --- WMMA fp32 GEMM: C = A@W (+ A2@W2) + bias, optional ReLU -----------------
// One wave computes one 16x16 output tile with a K-loop of 16x16x4 f32 WMMAs.
// blockDim = 128 (4 waves -> 64 rows per block). KP multiple of 4 (we use 16).
// VGPR layouts per CDNA5 ISA 7.12.2:
//   A 16x4 : lanes 0-15 row M, v0=K0 v1=K1 ; lanes 16-31 v0=K2 v1=K3
//   B 4x16 : lanes 0-15 col N, v0=K0 v1=K1 ; lanes 16-31 v0=K2 v1=K3
//   C 16x16: vgpr r -> (M=r, lanes 0-15) / (M=r+8, lanes 16-31)
__global__ void k_wmma_gemm(const float* __restrict__ A,  const float* __restrict__ W,
                            const float* __restrict__ A2, const float* __restrict__ W2,
                            const float* __restrict__ bias, float* __restrict__ C,
                            int KP, int ldA, int ldW, int ldC, int doRelu) {
    const int lane  = threadIdx.x & 31;
    const int wave  = threadIdx.x >> 5;
    const int row16 = lane & 15;
    const int hi    = lane >> 4;                      // 0: K+0/1, 1: K+2/3
    const int m0    = (blockIdx.x * 4 + wave) * 16;
    const int n0    = blockIdx.y * 16;

    v8f acc = {0.f, 0.f, 0.f, 0.f, 0.f, 0.f, 0.f, 0.f};

    const float* arow = A + (size_t)(m0 + row16) * ldA + 2 * hi;
    const float* wcol = W + (size_t)(2 * hi) * ldW + n0 + row16;
    for (int k = 0; k < KP; k += 4) {
        v2f a = *(const v2f*)(arow + k);              // 8B aligned (even k, even ld)
        v2f b;
        b.x = wcol[(size_t)k * ldW];
        b.y = wcol[(size_t)(k + 1) * ldW];
        acc = __builtin_amdgcn_wmma_f32_16x16x4_f32(false, a, false, b,
                                                    (short)0, acc, false, false);
    }
    if (A2 != nullptr) {                              // second product: agg @ Wn
        const float* arow2 = A2 + (size_t)(m0 + row16) * ldA + 2 * hi;
        const float* wcol2 = W2 + (size_t)(2 * hi) * ldW + n0 + row16;
        for (int k = 0; k < KP; k += 4) {
            v2f a = *(const v2f*)(arow2 + k);
            v2f b;
            b.x = wcol2[(size_t)k * ldW];
            b.y = wcol2[(size_t)(k + 1) * ldW];
            acc = __builtin_amdgcn_wmma_f32_16x16x4_f32(false, a, false, b,
                                                        (short)0, acc, false, false);
        }
    }

    float bv = bias[n0 + row16];
    #pragma unroll
    for (int r = 0; r < 8; ++r) {
        float v = acc[r] + bv;
        if (doRelu) v = fmaxf(v, 0.0f);
        C[(size_t)(m0 + r + 8 * hi) * ldC + n0 + row16] = v;
    }
}

// ---------------------------------------------------------------------------
extern "C" void kernel_launch(void* const* d_in, const int* in_sizes, int n_in,
                              void* d_out, int out_size, void* d_ws, size_t ws_size,
                              hipStream_t stream) {
    // setup_inputs() dict order:
    const float* x      = (const float*)d_in[0];   // [N,128]
    const int*   ei     = (const int*)  d_in[1];   // [2,E]
    const float* Wp0    = (const float*)d_in[2];   // [128,128]
    const float* bp0    = (const float*)d_in[3];   // [128]
    const float* Ws0    = (const float*)d_in[4];   // [128,70]
    const float* Wn0    = (const float*)d_in[5];   // [128,70]
    const float* b0     = (const float*)d_in[6];   // [70]
    const float* Wp_mid = (const float*)d_in[7];   // [3,70,70]
    const float* bp_mid = (const float*)d_in[8];   // [3,70]
    const float* Ws_mid = (const float*)d_in[9];   // [3,70,70]
    const float* Wn_mid = (const float*)d_in[10];  // [3,70,70]
    const float* b_mid  = (const float*)d_in[11];  // [3,70]
    const float* Wp4    = (const float*)d_in[12];  // [70,70]
    const float* bp4    = (const float*)d_in[13];  // [70]
    const float* Ws4    = (const float*)d_in[14];  // [70,64]
    const float* Wn4    = (const float*)d_in[15];  // [70,64]
    const float* b4     = (const float*)d_in[16];  // [64]

    const int Nn = in_sizes[0] / 128;              // 65536
    const int E  = in_sizes[1] / 2;                // 1048576
    const int* src = ei;
    const int* dst = ei + E;

    // workspace: 4 feature buffers [N x 128] + padded weight/bias staging
    float* ws  = (float*)d_ws;
    size_t per = (size_t)Nn * LDH;
    float* hA  = ws;
    float* hp  = ws + per;
    float* agg = ws + 2 * per;
    float* hB  = ws + 3 * per;
    float* Wpp = ws + 4 * per;        // up to 128x128
    float* Wsp = Wpp + 128 * 128;
    float* Wnp = Wsp + 128 * 128;
    float* bpp = Wnp + 128 * 128;     // 128
    float* bop = bpp + 128;           // 128

    auto padm = [&](const float* s, float* d, int rows, int cols, int RP, int CP) {
        int total = RP * CP;
        k_pad<<<(total + 255) / 256, 256, 0, stream>>>(s, d, rows, cols, RP, CP);
    };
    auto zero = [&](float* p, int n) {
        k_zero<<<(n + 255) / 256, 256, 0, stream>>>(p, n);
    };
    auto gemm = [&](const float* A, const float* W, const float* A2, const float* W2,
                    const float* bias, float* C, int KP, int ldA, int ldW, int ldC,
                    int NP, int relu) {
        dim3 grid(Nn / 64, NP / 16);
        k_wmma_gemm<<<grid, 128, 0, stream>>>(A, W, A2, W2, bias, C,
                                              KP, ldA, ldW, ldC, relu);
    };
    auto scatter = [&](const float* hpool, float* aggp, int ncols) {
        long t = (long)E * 32;
        k_scatter_max<<<(unsigned)((t + 255) / 256), 256, 0, stream>>>(
            hpool, src, dst, aggp, E, ncols);
    };

    // h = normalize(x)
    k_normalize<<<Nn / 8, 256, 0, stream>>>(x, hA, Nn);

    float* h_in  = hA;
    float* h_out = hB;

    // ---- layer 0: 128 -> 70 (pool dim 128) ----
    padm(Wp0, Wpp, 128, 128, 128, 128);  padm(bp0, bpp, 1, 128, 1, 128);
    padm(Ws0, Wsp, 128, 70, 128, 80);    padm(Wn0, Wnp, 128, 70, 128, 80);
    padm(b0,  bop, 1, 70, 1, 80);
    gemm(h_in, Wpp, nullptr, nullptr, bpp, hp, 128, LDH, 128, LDH, 128, 1);
    zero(agg, (int)per);
    scatter(hp, agg, 128);
    gemm(h_in, Wsp, agg, Wnp, bop, h_out, 128, LDH, 80, LDH, 80, 1);
    { float* t = h_in; h_in = h_out; h_out = t; }

    // ---- layers 1-3: 70 -> 70 (K padded 70 -> 80, zero-filled) ----
    for (int i = 0; i < 3; ++i) {
        const float* Wp = Wp_mid + (size_t)i * 70 * 70;
        const float* bp = bp_mid + (size_t)i * 70;
        const float* Wsl = Ws_mid + (size_t)i * 70 * 70;
        const float* Wnl = Wn_mid + (size_t)i * 70 * 70;
        const float* bl  = b_mid  + (size_t)i * 70;
        padm(Wp, Wpp, 70, 70, 80, 80);   padm(bp, bpp, 1, 70, 1, 80);
        padm(Wsl, Wsp, 70, 70, 80, 80);  padm(Wnl, Wnp, 70, 70, 80, 80);
        padm(bl, bop, 1, 70, 1, 80);
        gemm(h_in, Wpp, nullptr, nullptr, bpp, hp, 80, LDH, 80, LDH, 80, 1);
        zero(agg, (int)per);
        scatter(hp, agg, 80);
        gemm(h_in, Wsp, agg, Wnp, bop, h_out, 80, LDH, 80, LDH, 80, 1);
        { float* t = h_in; h_in = h_out; h_out = t; }
    }

    // ---- layer 4: 70 -> 64, no activation, write straight to d_out ----
    padm(Wp4, Wpp, 70, 70, 80, 80);      padm(bp4, bpp, 1, 70, 1, 80);
    padm(Ws4, Wsp, 70, 64, 80, 64);      padm(Wn4, Wnp, 70, 64, 80, 64);
    padm(b4,  bop, 1, 64, 1, 64);
    gemm(h_in, Wpp, nullptr, nullptr, bpp, hp, 80, LDH, 80, LDH, 80, 1);
    zero(agg, (int)per);
    scatter(hp, agg, 80);
    gemm(h_in, Wsp, agg, Wnp, bop, (float*)d_out, 80, LDH, 64, 64, 64, 0);
}